// GMSA_56667798503670
// MI455X (gfx1250) — compile-verified
//
#include <hip/hip_runtime.h>

typedef __attribute__((ext_vector_type(2))) float v2f;
typedef __attribute__((ext_vector_type(8))) float v8f;

#define D_FEAT 128
#define LDS_STRIDE 136   // 2*136 mod 64 == 16 -> half-wave bank groups disjoint

// ---------------------------------------------------------------------------
// Degree accumulation: one atomic per edge endpoint.
// ---------------------------------------------------------------------------
__global__ void deg_kernel(const int* __restrict__ idx, float* __restrict__ deg, int e) {
    int i = blockIdx.x * blockDim.x + threadIdx.x;
    if (i < e) atomicAdd(&deg[idx[i]], 1.0f);
}

// deg -> norm in place: norm = deg>0 ? rsqrt(deg) : 0   (covers all 4 arrays in one pass)
__global__ void deg_to_norm_kernel(float* __restrict__ d, int n) {
    int i = blockIdx.x * blockDim.x + threadIdx.x;
    if (i < n) {
        float v = d[i];
        d[i] = (v > 0.0f) ? rsqrtf(v) : 0.0f;
    }
}

// ---------------------------------------------------------------------------
// Edge scatter: one wave32 per edge. Lane l handles floats [4l,4l+4) of the
// 128-float feature row: coalesced 512B gather (scaled by src norm) and a
// 512B row of f32 atomic adds into agg[dst]. Feature tables are L2-resident.
// ---------------------------------------------------------------------------
__global__ void scatter_kernel(const float* __restrict__ feat,
                               const int* __restrict__ src,
                               const int* __restrict__ dst,
                               const float* __restrict__ norm_src,
                               float* __restrict__ agg, int e) {
    long long gid = (long long)blockIdx.x * blockDim.x + threadIdx.x;
    long long total = (long long)e * 32;
    long long stride = (long long)gridDim.x * blockDim.x;
    for (; gid < total; gid += stride) {
        int edge = (int)(gid >> 5);
        int c = ((int)gid & 31) * 4;
        int s = src[edge];
        int d = dst[edge];
        float ns = norm_src[s];
        float4 f = *(const float4*)&feat[(size_t)s * D_FEAT + c];
        float* base = &agg[(size_t)d * D_FEAT + c];
        atomicAdd(base + 0, f.x * ns);
        atomicAdd(base + 1, f.y * ns);
        atomicAdd(base + 2, f.z * ns);
        atomicAdd(base + 3, f.w * ns);
    }
}

// ---------------------------------------------------------------------------
// Fused  Y = relu( (X * rowscale) @ W + bias )  via V_WMMA_F32_16X16X4_F32.
// Block = 256 threads = 8 waves; W (128x128 f32) is staged once per block
// into LDS (row stride 136 floats -> conflict-free half-wave B reads) and
// shared by all 8 waves. Each wave owns a 16x128 output stripe.
//   A 16x4 f32: lane l -> row m=l&15, k-half kh=(l>>4)*2, VGPR0/1 = k,k+1.
//   B  4x16 f32 mirrors A (lane l -> col n=l&15), read from LDS.
//   C/D 16x16 f32: VGPR v -> row v + (l>>4)*8, col l&15.
// rowscale (dst-degree norm) folded into the A load; bias+ReLU in epilogue.
// ---------------------------------------------------------------------------
__global__ __launch_bounds__(256) void gcn_gemm_relu_kernel(
        const float* __restrict__ X, const float* __restrict__ rowscale,
        const float* __restrict__ W, const float* __restrict__ bias,
        float* __restrict__ Y, int nrows) {
    __shared__ float Wlds[D_FEAT * LDS_STRIDE];   // 69632 B, < 320 KB WGP pool

    // cooperative stage of W: 2 threads per row, float4 copies (256 B/thread)
    {
        int r    = threadIdx.x >> 1;
        int half = (threadIdx.x & 1) * 64;
        const float4* srcp = (const float4*)&W[(size_t)r * D_FEAT + half];
        float4*       dstp = (float4*)&Wlds[r * LDS_STRIDE + half];
#pragma unroll
        for (int i = 0; i < 16; i++) dstp[i] = srcp[i];
    }
    __syncthreads();

    int wave = threadIdx.x >> 5;
    int lane = threadIdx.x & 31;
    int row0 = (blockIdx.x * 8 + wave) * 16;
    if (row0 >= nrows) return;              // wave-uniform, after barrier: EXEC all-1s

    int m  = lane & 15;
    int kh = (lane >> 4) * 2;

    float rs = rowscale[row0 + m];
    const float* Xrow = X + (size_t)(row0 + m) * D_FEAT;

    v8f acc[8] = {};
    for (int k0 = 0; k0 < D_FEAT; k0 += 4) {
        v2f a;
        a.x = Xrow[k0 + kh + 0] * rs;
        a.y = Xrow[k0 + kh + 1] * rs;
        const float* wrow0 = &Wlds[(k0 + kh + 0) * LDS_STRIDE];
        const float* wrow1 = &Wlds[(k0 + kh + 1) * LDS_STRIDE];
#pragma unroll
        for (int nt = 0; nt < 8; nt++) {
            int n = nt * 16 + (lane & 15);
            v2f b;
            b.x = wrow0[n];
            b.y = wrow1[n];
            acc[nt] = __builtin_amdgcn_wmma_f32_16x16x4_f32(
                false, a, false, b, (short)0, acc[nt], false, false);
        }
    }

    int rbase = row0 + (lane >> 4) * 8;
    int ncol  = lane & 15;
#pragma unroll
    for (int nt = 0; nt < 8; nt++) {
        int col = nt * 16 + ncol;
        float bv = bias[col];
#pragma unroll
        for (int v = 0; v < 8; v++) {
            float r = acc[nt][v] + bv;
            Y[(size_t)(rbase + v) * D_FEAT + col] = r > 0.0f ? r : 0.0f;
        }
    }
}

// ---------------------------------------------------------------------------
// Per-node head attention (Q=K=V=h): scores[h]=sum_d h[h*16+d]^2 / 4,
// softmax over 8 heads, out = score[head]*h. One thread per node, float4 IO.
// ---------------------------------------------------------------------------
__global__ void attn_kernel(const float* __restrict__ h, float* __restrict__ out, int n) {
    int i = blockIdx.x * blockDim.x + threadIdx.x;
    if (i >= n) return;
    const float* row = h + (size_t)i * D_FEAT;
    float s[8];
    float mx = -1e30f;
#pragma unroll
    for (int hh = 0; hh < 8; hh++) {
        float acc = 0.0f;
#pragma unroll
        for (int d = 0; d < 16; d += 4) {
            float4 q = *(const float4*)&row[hh * 16 + d];
            acc += q.x * q.x + q.y * q.y + q.z * q.z + q.w * q.w;
        }
        s[hh] = acc * 0.25f;                 // / sqrt(HD=16)
        mx = fmaxf(mx, s[hh]);
    }
    float denom = 0.0f;
#pragma unroll
    for (int hh = 0; hh < 8; hh++) { s[hh] = expf(s[hh] - mx); denom += s[hh]; }
    float inv = 1.0f / denom;
    float* orow = out + (size_t)i * D_FEAT;
#pragma unroll
    for (int hh = 0; hh < 8; hh++) {
        float w = s[hh] * inv;
#pragma unroll
        for (int d = 0; d < 16; d += 4) {
            float4 q = *(const float4*)&row[hh * 16 + d];
            float4 r = make_float4(q.x * w, q.y * w, q.z * w, q.w * w);
            *(float4*)&orow[hh * 16 + d] = r;
        }
    }
}

// ---------------------------------------------------------------------------
// In-place LayerNorm over 128 features, one thread per node.
// ---------------------------------------------------------------------------
__global__ void layernorm_kernel(float* __restrict__ out,
                                 const float* __restrict__ gamma,
                                 const float* __restrict__ beta, int n) {
    int i = blockIdx.x * blockDim.x + threadIdx.x;
    if (i >= n) return;
    float* row = out + (size_t)i * D_FEAT;
    float sum = 0.0f, sumsq = 0.0f;
    for (int d = 0; d < D_FEAT; d += 4) {
        float4 x = *(const float4*)&row[d];
        sum   += x.x + x.y + x.z + x.w;
        sumsq += x.x * x.x + x.y * x.y + x.z * x.z + x.w * x.w;
    }
    float mu  = sum * (1.0f / D_FEAT);
    float var = sumsq * (1.0f / D_FEAT) - mu * mu;
    float inv = rsqrtf(var + 1e-5f);
    for (int d = 0; d < D_FEAT; d += 4) {
        float4 x = *(const float4*)&row[d];
        float4 g = *(const float4*)&gamma[d];
        float4 b = *(const float4*)&beta[d];
        float4 r = make_float4((x.x - mu) * inv * g.x + b.x,
                               (x.y - mu) * inv * g.y + b.y,
                               (x.z - mu) * inv * g.z + b.z,
                               (x.w - mu) * inv * g.w + b.w);
        *(float4*)&row[d] = r;
    }
}

// ---------------------------------------------------------------------------
extern "C" void kernel_launch(void* const* d_in, const int* in_sizes, int n_in,
                              void* d_out, int out_size, void* d_ws, size_t ws_size,
                              hipStream_t stream) {
    const float* feat  = (const float*)d_in[0];
    const int*   src1  = (const int*)d_in[1];
    const int*   dst1  = (const int*)d_in[2];
    const int*   src2  = (const int*)d_in[3];
    const int*   dst2  = (const int*)d_in[4];
    const float* W1    = (const float*)d_in[5];
    const float* b1    = (const float*)d_in[6];
    const float* W2    = (const float*)d_in[7];
    const float* b2    = (const float*)d_in[8];
    const float* gamma = (const float*)d_in[9];
    const float* beta  = (const float*)d_in[10];
    float* out = (float*)d_out;

    const int N = in_sizes[0] / D_FEAT;
    const int E = in_sizes[1];

    // workspace layout: 4 norm arrays (contiguous) + 2 node-feature buffers
    float* normO1 = (float*)d_ws;              // out-deg norm, graph 1
    float* normI1 = normO1 + N;                // in-deg  norm, graph 1
    float* normO2 = normI1 + N;                // out-deg norm, graph 2
    float* normI2 = normO2 + N;                // in-deg  norm, graph 2
    float* bufA   = normI2 + N;                // N*128
    float* bufB   = bufA + (size_t)N * D_FEAT; // N*128

    const int TB = 256;
    int degBlocks  = (E + TB - 1) / TB;
    int nodeBlocks = (N + TB - 1) / TB;
    long long scatterThreads = (long long)E * 32;
    int scatterBlocks = (int)((scatterThreads + TB - 1) / TB);
    int stripes = N / 16;                      // N divisible by 16 (50000 = 3125*16)
    int gemmBlocks = (stripes + 7) / 8;        // 8 waves (16-row stripes) per block

    // ---- degrees -> norms (4 arrays, one fused transform) ----
    hipMemsetAsync(normO1, 0, (size_t)4 * N * sizeof(float), stream);
    deg_kernel<<<degBlocks, TB, 0, stream>>>(src1, normO1, E);
    deg_kernel<<<degBlocks, TB, 0, stream>>>(dst1, normI1, E);
    deg_kernel<<<degBlocks, TB, 0, stream>>>(src2, normO2, E);
    deg_kernel<<<degBlocks, TB, 0, stream>>>(dst2, normI2, E);
    deg_to_norm_kernel<<<(4 * N + TB - 1) / TB, TB, 0, stream>>>(normO1, 4 * N);

    // ---- GraphConv 1: scatter feat -> bufA, WMMA GEMM -> bufB (= h) ----
    hipMemsetAsync(bufA, 0, (size_t)N * D_FEAT * sizeof(float), stream);
    scatter_kernel<<<scatterBlocks, TB, 0, stream>>>(feat, src1, dst1, normO1, bufA, E);
    gcn_gemm_relu_kernel<<<gemmBlocks, 256, 0, stream>>>(bufA, normI1, W1, b1, bufB, N);

    // ---- head attention: bufB (h) -> bufA (attended) ----
    attn_kernel<<<nodeBlocks, TB, 0, stream>>>(bufB, bufA, N);

    // ---- GraphConv 2: scatter bufA -> bufB (re-zeroed), WMMA GEMM -> out ----
    hipMemsetAsync(bufB, 0, (size_t)N * D_FEAT * sizeof(float), stream);
    scatter_kernel<<<scatterBlocks, TB, 0, stream>>>(bufA, src2, dst2, normO2, bufB, E);
    gcn_gemm_relu_kernel<<<gemmBlocks, 256, 0, stream>>>(bufB, normI2, W2, b2, out, N);

    // ---- LayerNorm in place on d_out ----
    layernorm_kernel<<<nodeBlocks, TB, 0, stream>>>(out, gamma, beta, N);
}